// PromptEncoderClickmap_58334245814865
// MI455X (gfx1250) — compile-verified
//
#include <hip/hip_runtime.h>

#define VOL 192
#define RAD 8
#define KW  17          // 2*RAD+1
#define NV  (KW*KW*KW)  // 4913
#define NUM_PTS 16      // points per batch (reference: bs=4, num_pts=16)
#define MAXP 256

typedef __attribute__((ext_vector_type(2))) float v2f;
typedef __attribute__((ext_vector_type(4))) float v4f;
typedef __attribute__((ext_vector_type(8))) float v8f;

// ---------------------------------------------------------------------------
// Kernel 1: streaming zero-fill of the whole output (click_map + embeddings).
// 340 MB write-only -> b128 non-temporal stores, grid-stride.
// ---------------------------------------------------------------------------
__global__ void pe_zero_fill(float* __restrict__ out, long n) {
    const long stride = (long)gridDim.x * blockDim.x;
    long i = (long)blockIdx.x * blockDim.x + threadIdx.x;
    const long n4 = n >> 2;
    v4f z = {0.0f, 0.0f, 0.0f, 0.0f};
    v4f* __restrict__ o4 = (v4f*)out;
    for (long k = i; k < n4; k += stride)
        __builtin_nontemporal_store(z, o4 + k);
    const long tail = n & 3;
    if (i < tail) out[(n4 << 2) + i] = 0.0f;
}

// ---------------------------------------------------------------------------
// Kernel 2: build the 17^3 Gaussian table with one wave using WMMA.
// D = gy (16x1, in A K=0) x gx (1x16, in B K=0)  via V_WMMA_F32_16X16X4_F32.
// Index 16 row/col/plane patched via g(16)==g(0) symmetry (bit-exact: (−8)²=8²).
// Reference zeroing: value = 0 where rx²+ry²+rz² >= 128 (h < eps*h.max()).
// ---------------------------------------------------------------------------
__global__ void pe_build_table_wmma(float* __restrict__ table) {
    const int lane = threadIdx.x & 31;
    const bool lo  = lane < 16;
    const int  n   = lane & 15;

    const float rn = (float)(n - RAD);
    const float g  = expf(-(rn * rn) * 0.125f);   // exp(-r^2 / (2*sigma^2)), sigma=2

    // A 16x4 f32: VGPR0 = K=0 (lanes 0-15, M=lane), K=2 (lanes 16-31); VGPR1 = K=1/K=3.
    // Only K=0 carries data -> rank-1 outer product.
    v2f a; a.x = lo ? g : 0.0f; a.y = 0.0f;
    v2f b = a;                                    // gx == gy (same 1-D Gaussian)
    v8f c = {};
    v8f d = __builtin_amdgcn_wmma_f32_16x16x4_f32(
        /*neg_a=*/false, a, /*neg_b=*/false, b,
        /*c_mod=*/(short)0, c, /*reuse_a=*/false, /*reuse_b=*/false);

    // D layout: lane L, VGPR v -> M = v + (L>=16 ? 8 : 0), N = L%16.
    for (int dz = 0; dz < KW; ++dz) {
        const int   rz  = dz - RAD;
        const int   rz2 = rz * rz;
        const float fz  = (float)rz;
        const float gz  = expf(-(fz * fz) * 0.125f);

        #pragma unroll
        for (int v = 0; v < 8; ++v) {
            const int M  = v + (lo ? 0 : 8);
            const int ry = M - RAD, rx = n - RAD;
            const int r2 = rz2 + ry * ry + rx * rx;
            const float val = (r2 >= 128) ? 0.0f : gz * d[v];
            table[(dz * KW + M) * KW + n] = val;
        }
        // Row M=16 (== row 0 values): lanes 0..15 hold D[0][n] in d[0].
        if (lo) {
            const int rx = n - RAD;
            const int r2 = rz2 + 64 + rx * rx;
            table[(dz * KW + 16) * KW + n] = (r2 >= 128) ? 0.0f : gz * d[0];
        }
        // Col N=16 (== col 0 values): lanes with n==0 hold D[M][0].
        if (n == 0) {
            #pragma unroll
            for (int v = 0; v < 8; ++v) {
                const int M  = v + (lo ? 0 : 8);
                const int ry = M - RAD;
                const int r2 = rz2 + ry * ry + 64;
                table[(dz * KW + M) * KW + 16] = (r2 >= 128) ? 0.0f : gz * d[v];
            }
            if (lo) {
                // Corner (16,16): r2 = rz2 + 128 >= 128 -> always zeroed.
                table[(dz * KW + 16) * KW + 16] = 0.0f;
            }
        }
    }
}

// ---------------------------------------------------------------------------
// Kernel 3: scatter Gaussian blocks with last-writer-wins semantics.
// One block per point; each voxel checks all LATER points of the same
// (batch, class) for Chebyshev-<=RAD coverage (they own the voxel, including
// their zero corners), reproducing the reference's sequential overwrite.
// ---------------------------------------------------------------------------
__global__ void pe_scatter(const float* __restrict__ coords,
                           const int*   __restrict__ labels,
                           const float* __restrict__ table,
                           float* __restrict__ out,
                           int total_pts) {
    __shared__ int sb[MAXP], sx[MAXP], sy[MAXP], sz[MAXP], sc[MAXP];
    const int tid = threadIdx.x;
    if (tid < total_pts) {
        sb[tid] = tid / NUM_PTS;
        sx[tid] = (int)coords[tid * 3 + 0];   // x (W)
        sy[tid] = (int)coords[tid * 3 + 1];   // y (H)
        sz[tid] = (int)coords[tid * 3 + 2];   // z (D)
        const int l = labels[tid];
        sc[tid] = (l == 1) ? 0 : ((l == 0) ? 1 : -1);   // 0=pos ch, 1=neg ch, -1=skip
    }
    __syncthreads();

    const int i   = blockIdx.x;
    const int cls = sc[i];
    if (cls < 0) return;
    const int b = sb[i], cx = sx[i], cy = sy[i], cz = sz[i];

    for (int idx = tid; idx < NV; idx += blockDim.x) {
        const int dx = idx % KW;
        const int t  = idx / KW;
        const int dy = t % KW;
        const int dz = t / KW;
        const int px = cx - RAD + dx;
        const int py = cy - RAD + dy;
        const int pz = cz - RAD + dz;
        if ((unsigned)px >= (unsigned)VOL ||
            (unsigned)py >= (unsigned)VOL ||
            (unsigned)pz >= (unsigned)VOL) continue;

        // Does a later point of the same (batch, class) cover this voxel?
        bool covered = false;
        for (int j = i + 1; j < total_pts; ++j) {
            if (sb[j] != b || sc[j] != cls) continue;
            int ax = px - sx[j]; ax = ax < 0 ? -ax : ax;
            int ay = py - sy[j]; ay = ay < 0 ? -ay : ay;
            int az = pz - sz[j]; az = az < 0 ? -az : az;
            if (ax <= RAD && ay <= RAD && az <= RAD) { covered = true; break; }
        }
        if (covered) continue;

        const size_t o =
            ((((size_t)b * 2 + (size_t)cls) * VOL + (size_t)pz) * VOL +
             (size_t)py) * VOL + (size_t)px;
        out[o] = table[idx];   // max==1 & min==0 make normalization an identity
    }
}

// ---------------------------------------------------------------------------
extern "C" void kernel_launch(void* const* d_in, const int* in_sizes, int n_in,
                              void* d_out, int out_size, void* d_ws, size_t ws_size,
                              hipStream_t stream) {
    const float* coords = (const float*)d_in[0];   // (bs, num_pts, 3) f32
    const int*   labels = (const int*)d_in[1];     // (bs, num_pts)    i32
    float* out   = (float*)d_out;                  // click_map ++ dense_embeddings
    float* table = (float*)d_ws;                   // 17^3 floats (~19.6 KB)

    const int total_pts = in_sizes[1];             // bs * num_pts = 64

    // 1) zero everything (click_map background + all-zero dense_embeddings)
    pe_zero_fill<<<2048, 256, 0, stream>>>(out, (long)out_size);

    // 2) build Gaussian table with one wave (WMMA outer product)
    pe_build_table_wmma<<<1, 32, 0, stream>>>(table);

    // 3) scatter blocks, sequential-overwrite semantics via later-point checks
    pe_scatter<<<total_pts, 256, 0, stream>>>(coords, labels, table, out, total_pts);
}